// TGGuidedSelfAttention_46231027974629
// MI455X (gfx1250) — compile-verified
//
#include <hip/hip_runtime.h>
#include <math.h>

typedef __attribute__((ext_vector_type(2))) float v2f;
typedef __attribute__((ext_vector_type(8))) float v8f;

#define WMMA_F32(a, b, c) \
  __builtin_amdgcn_wmma_f32_16x16x4_f32(false, (a), false, (b), (short)0, (c), false, false)

constexpr int B_   = 8;
constexpr int N_   = 1025;
constexpr int D_   = 768;
constexpr int H_   = 12;
constexpr int HD_  = 64;
constexpr int P_   = 1024;
constexpr int NPAD = 1040;          // 65 * 16
constexpr int M_   = B_ * N_;       // 8200 rows of x
constexpr int NT_  = NPAD / 16;     // 65 j/i tiles
constexpr float SCALE_ = 0.125f;    // 64^-0.5
constexpr float EPS_   = 1e-6f;

// ---------------------------------------------------------------- pw = t / (sum t + eps)
__global__ void pw_kernel(const float* __restrict__ t, float* __restrict__ pw) {
  __shared__ float red[256];
  const int b = blockIdx.x, tid = threadIdx.x;
  float s = 0.f;
  for (int j = tid; j < P_; j += 256) s += t[b * P_ + j];
  red[tid] = s;
  __syncthreads();
  for (int off = 128; off > 0; off >>= 1) {
    if (tid < off) red[tid] += red[tid + off];
    __syncthreads();
  }
  const float inv = 1.0f / (red[0] + EPS_);
  for (int j = tid; j < P_; j += 256) pw[b * P_ + j] = t[b * P_ + j] * inv;
}

// ---------------------------------------------------------------- zero pad rows of Q/K/V
__global__ void zeropad_kernel(float* __restrict__ Q, float* __restrict__ K,
                               float* __restrict__ V) {
  const int per = B_ * H_ * (NPAD - N_) * HD_;    // 92160
  int idx = blockIdx.x * 256 + threadIdx.x;
  if (idx >= per) return;
  int hd = idx % HD_;
  int r  = (idx / HD_) % (NPAD - N_);
  int bh = idx / (HD_ * (NPAD - N_));
  size_t off = ((size_t)bh * NPAD + (N_ + r)) * HD_ + hd;
  Q[off] = 0.f; K[off] = 0.f; V[off] = 0.f;
}

// ---------------------------------------------------------------- QKV GEMM (M=8200, N=2304, K=768)
// block = 4 waves, each wave a 16x64 tile; grid = (2304/256, ceil(8200/16))
__global__ __launch_bounds__(128) void qkv_kernel(
    const float* __restrict__ X, const float* __restrict__ W,
    const float* __restrict__ bias, float* __restrict__ Q,
    float* __restrict__ K, float* __restrict__ V) {
  const int lane = threadIdx.x & 31;
  const int wave = threadIdx.x >> 5;
  const int hi   = lane >> 4;
  const int l16  = lane & 15;
  const int m0   = blockIdx.y * 16;
  const int n0   = blockIdx.x * 256 + wave * 64;

  int mrow = m0 + l16;
  if (mrow >= M_) mrow = M_ - 1;                 // clamp; stores are guarded
  const float* xrow = X + (size_t)mrow * D_;
  const float* w0 = W + (size_t)(n0 + l16) * D_;
  const float* w1 = w0 + (size_t)16 * D_;
  const float* w2 = w0 + (size_t)32 * D_;
  const float* w3 = w0 + (size_t)48 * D_;

  v8f acc0 = {}, acc1 = {}, acc2 = {}, acc3 = {};
  for (int k = 0; k < D_; k += 4) {
    const int kk = k + 2 * hi;
    v2f a  = *(const v2f*)(xrow + kk);
    v2f b0 = *(const v2f*)(w0 + kk);
    v2f b1 = *(const v2f*)(w1 + kk);
    v2f b2 = *(const v2f*)(w2 + kk);
    v2f b3 = *(const v2f*)(w3 + kk);
    acc0 = WMMA_F32(a, b0, acc0);
    acc1 = WMMA_F32(a, b1, acc1);
    acc2 = WMMA_F32(a, b2, acc2);
    acc3 = WMMA_F32(a, b3, acc3);
  }

#pragma unroll
  for (int t = 0; t < 4; ++t) {
    v8f acc = (t == 0) ? acc0 : (t == 1) ? acc1 : (t == 2) ? acc2 : acc3;
    const int n   = n0 + t * 16 + l16;
    const int which = n / D_;
    const int d   = n % D_;
    const int h   = d >> 6, hd = d & 63;
    const float bv = bias[n];
    float* dst = (which == 0) ? Q : (which == 1) ? K : V;
#pragma unroll
    for (int r = 0; r < 8; ++r) {
      const int m = m0 + r + 8 * hi;
      if (m < M_) {
        const int b = m / N_;
        const int i = m - b * N_;
        dst[(((size_t)b * H_ + h) * NPAD + i) * HD_ + hd] = acc[r] + bv;
      }
    }
  }
}

// ---------------------------------------------------------------- fused masked flash attention
// one wave per (b,h, 16-query tile); 8 waves / block; grid = B*H*65/8 = 780
__global__ __launch_bounds__(256) void attn_kernel(
    const float* __restrict__ Q, const float* __restrict__ Kb,
    const float* __restrict__ Vb, const float* __restrict__ graph,
    const float* __restrict__ pw, float* __restrict__ ctx) {
  constexpr int LSTR = 18;                       // LDS row stride (floats)
  __shared__ float ptile[8 * 16 * LSTR];
  const int lane = threadIdx.x & 31;
  const int wave = threadIdx.x >> 5;
  const int hi   = lane >> 4;
  const int l16  = lane & 15;

  const int gid = blockIdx.x * 8 + wave;         // 0..6239
  const int bh  = gid / NT_;
  const int it  = gid - bh * NT_;
  const int b   = bh / H_;
  const int h   = bh - b * H_;
  const int i0  = it * 16;

  const float* qbase = Q  + (size_t)bh * NPAD * HD_;
  const float* kbase = Kb + (size_t)bh * NPAD * HD_;
  const float* vbase = Vb + (size_t)bh * NPAD * HD_;
  float* lds = ptile + wave * 16 * LSTR;

  v8f acc0 = {}, acc1 = {}, acc2 = {}, acc3 = {};
  float rmax[8], rZ[8], rS[8];
#pragma unroll
  for (int r = 0; r < 8; ++r) { rmax[r] = -3.0e38f; rZ[r] = 0.f; rS[r] = 0.f; }

  const float* qrow = qbase + (size_t)(i0 + l16) * HD_;

  for (int jt = 0; jt < NT_; ++jt) {
    const int j0 = jt * 16;
    // ---- S = (Q K^T) for a 16x16 tile, K-dim = 64 in steps of 4
    const float* krow = kbase + (size_t)(j0 + l16) * HD_;
    v8f s = {};
#pragma unroll
    for (int k = 0; k < HD_; k += 4) {
      v2f a  = *(const v2f*)(qrow + k + 2 * hi);
      v2f bf = *(const v2f*)(krow + k + 2 * hi);
      s = WMMA_F32(a, bf, s);
    }

    const int j    = j0 + l16;                   // column owned by this lane
    const bool jpad = (j >= N_);
    const int jc   = jpad ? (N_ - 1) : j;        // clamped for safe loads

    // ---- per-row online softmax + mask, rows r (lanes<16) / r+8 (lanes>=16)
#pragma unroll
    for (int r = 0; r < 8; ++r) {
      float x = jpad ? -3.0e38f : s[r] * SCALE_; // select kills pad garbage/NaN
      // row max across the 16 lanes of this half-wave
      float v = x;
#pragma unroll
      for (int msk = 1; msk < 16; msk <<= 1) v = fmaxf(v, __shfl_xor(v, msk, 32));
      const float nm = fmaxf(rmax[r], v);
      const float f  = __expf(rmax[r] - nm);
      rmax[r] = nm;
      const float e = __expf(x - nm);
      float esum = e;
#pragma unroll
      for (int msk = 1; msk < 16; msk <<= 1) esum += __shfl_xor(esum, msk, 32);
      rZ[r] = rZ[r] * f + esum;

      // mask m[b, iRow, j]
      const int iRow = i0 + r + 8 * hi;
      const int ic   = (iRow < N_) ? iRow : (N_ - 1);
      float mval;
      if (ic == 0) {
        mval = (jc == 0) ? 1.0f : pw[b * P_ + jc - 1];
      } else {
        mval = (jc == 0) ? pw[b * P_ + ic - 1]
                         : graph[((size_t)b * P_ + (ic - 1)) * P_ + (jc - 1)];
      }
      const float em = e * mval;
      float emsum = em;
#pragma unroll
      for (int msk = 1; msk < 16; msk <<= 1) emsum += __shfl_xor(emsum, msk, 32);
      rS[r] = rS[r] * f + emsum;

      // rescale accumulators for this row
      acc0[r] *= f; acc1[r] *= f; acc2[r] *= f; acc3[r] *= f;

      // stage e*m tile to LDS (C-layout -> A-layout transpose), stride 18
      lds[(r + 8 * hi) * LSTR + l16] = em;
    }
    asm volatile("s_wait_dscnt 0" ::: "memory");

    // ---- acc += P(16xj16) @ V(j16 x 64), 4 K-steps x 4 d-tiles
#pragma unroll
    for (int k0 = 0; k0 < 16; k0 += 4) {
      const int ar = l16 * LSTR + k0 + 2 * hi;
      v2f a; a.x = lds[ar]; a.y = lds[ar + 1];
      const float* vp = vbase + (size_t)(j0 + k0 + 2 * hi) * HD_;
      v2f b0, b1, b2, b3;
      b0.x = vp[ 0 + l16]; b0.y = vp[HD_ +  0 + l16];
      b1.x = vp[16 + l16]; b1.y = vp[HD_ + 16 + l16];
      b2.x = vp[32 + l16]; b2.y = vp[HD_ + 32 + l16];
      b3.x = vp[48 + l16]; b3.y = vp[HD_ + 48 + l16];
      acc0 = WMMA_F32(a, b0, acc0);
      acc1 = WMMA_F32(a, b1, acc1);
      acc2 = WMMA_F32(a, b2, acc2);
      acc3 = WMMA_F32(a, b3, acc3);
    }
  }

  // ---- exact renorm: out = acc / (S + eps * Z); store ctx[b, i, h*64 + d]
#pragma unroll
  for (int r = 0; r < 8; ++r) {
    const float inv = 1.0f / (rS[r] + EPS_ * rZ[r]);
    const int iRow = i0 + r + 8 * hi;
    if (iRow < N_) {
      const size_t base = ((size_t)(b * N_ + iRow)) * D_ + h * HD_;
      ctx[base +  0 + l16] = acc0[r] * inv;
      ctx[base + 16 + l16] = acc1[r] * inv;
      ctx[base + 32 + l16] = acc2[r] * inv;
      ctx[base + 48 + l16] = acc3[r] * inv;
    }
  }
}

// ---------------------------------------------------------------- output projection GEMM
// M=8200, N=768, K=768; grid = (768/256, ceil(8200/16)), 4 waves/block
__global__ __launch_bounds__(128) void proj_kernel(
    const float* __restrict__ X, const float* __restrict__ W,
    const float* __restrict__ bias, float* __restrict__ out) {
  const int lane = threadIdx.x & 31;
  const int wave = threadIdx.x >> 5;
  const int hi   = lane >> 4;
  const int l16  = lane & 15;
  const int m0   = blockIdx.y * 16;
  const int n0   = blockIdx.x * 256 + wave * 64;

  int mrow = m0 + l16;
  if (mrow >= M_) mrow = M_ - 1;
  const float* xrow = X + (size_t)mrow * D_;
  const float* w0 = W + (size_t)(n0 + l16) * D_;
  const float* w1 = w0 + (size_t)16 * D_;
  const float* w2 = w0 + (size_t)32 * D_;
  const float* w3 = w0 + (size_t)48 * D_;

  v8f acc0 = {}, acc1 = {}, acc2 = {}, acc3 = {};
  for (int k = 0; k < D_; k += 4) {
    const int kk = k + 2 * hi;
    v2f a  = *(const v2f*)(xrow + kk);
    v2f b0 = *(const v2f*)(w0 + kk);
    v2f b1 = *(const v2f*)(w1 + kk);
    v2f b2 = *(const v2f*)(w2 + kk);
    v2f b3 = *(const v2f*)(w3 + kk);
    acc0 = WMMA_F32(a, b0, acc0);
    acc1 = WMMA_F32(a, b1, acc1);
    acc2 = WMMA_F32(a, b2, acc2);
    acc3 = WMMA_F32(a, b3, acc3);
  }

#pragma unroll
  for (int t = 0; t < 4; ++t) {
    v8f acc = (t == 0) ? acc0 : (t == 1) ? acc1 : (t == 2) ? acc2 : acc3;
    const int n = n0 + t * 16 + l16;
    const float bv = bias[n];
#pragma unroll
    for (int r = 0; r < 8; ++r) {
      const int m = m0 + r + 8 * hi;
      if (m < M_) out[(size_t)m * D_ + n] = acc[r] + bv;
    }
  }
}

// ----------------------------------------------------------------
extern "C" void kernel_launch(void* const* d_in, const int* in_sizes, int n_in,
                              void* d_out, int out_size, void* d_ws, size_t ws_size,
                              hipStream_t stream) {
  const float* x      = (const float*)d_in[0];
  const float* graph  = (const float*)d_in[1];
  const float* transf = (const float*)d_in[2];
  const float* Wqkv   = (const float*)d_in[3];
  const float* bqkv   = (const float*)d_in[4];
  const float* Wproj  = (const float*)d_in[5];
  const float* bproj  = (const float*)d_in[6];
  float* out = (float*)d_out;

  float* ws = (float*)d_ws;
  float* pw = ws;                                  // 8192 floats
  float* Q  = pw + 8192;
  const size_t qsz = (size_t)B_ * H_ * NPAD * HD_; // 6,389,760 floats each
  float* K  = Q + qsz;
  float* V  = K + qsz;
  float* ctx = V + qsz;                            // 8200*768 floats

  pw_kernel<<<B_, 256, 0, stream>>>(transf, pw);
  const int padElems = B_ * H_ * (NPAD - N_) * HD_;
  zeropad_kernel<<<(padElems + 255) / 256, 256, 0, stream>>>(Q, K, V);
  qkv_kernel<<<dim3(2304 / 256, (M_ + 15) / 16), 128, 0, stream>>>(x, Wqkv, bqkv, Q, K, V);
  attn_kernel<<<(B_ * H_ * NT_) / 8, 256, 0, stream>>>(Q, K, V, graph, pw, ctx);
  proj_kernel<<<dim3(768 / 256, (M_ + 15) / 16), 128, 0, stream>>>(ctx, Wproj, bproj, out);
}